// HANLayer_39883066310777
// MI455X (gfx1250) — compile-verified
//
#include <hip/hip_runtime.h>
#include <math.h>

#define N_NODES 100000
#define N_EDGES 800000
#define IN_DIM  256
#define HEADS   8
#define DDIM    32
#define HD      256   // HEADS * DDIM
#define NEG_SLOPE 0.2f

typedef float v2f __attribute__((ext_vector_type(2)));
typedef float v8f __attribute__((ext_vector_type(8)));

// ---------------------------------------------------------------------------
// Kernel 1: feat = h @ W_fc  (100000x256 @ 256x256), fp32 WMMA 16x16x4.
// One wave per 16(M) x 64(N) strip; 4 accumulators; A fragment reused 4x.
// ---------------------------------------------------------------------------
__global__ __launch_bounds__(32)
void gemm_feat_wmma(const float* __restrict__ h,
                    const float* __restrict__ W,
                    float* __restrict__ feat) {
    const int m0 = blockIdx.x * 16;        // 6250 tiles in M
    const int n0 = blockIdx.y * 64;        // 4 strips in N
    const int lane  = threadIdx.x;         // 0..31 (wave32)
    const int l16   = lane & 15;
    const int lhalf = lane >> 4;           // 0 or 1

    v8f acc0 = {}; v8f acc1 = {}; v8f acc2 = {}; v8f acc3 = {};

    const float* arow = h + (size_t)(m0 + l16) * IN_DIM + 2 * lhalf;

    for (int k0 = 0; k0 < IN_DIM; k0 += 4) {
        // A fragment: row (m0+l16), cols k0 + 2*lhalf + {0,1}  (contiguous pair)
        v2f a = *(const v2f*)(arow + k0);

        // B fragments: rows k0 + 2*lhalf + {0,1}, col n0 + j*16 + l16
        const float* brow0 = W + (size_t)(k0 + 2 * lhalf + 0) * HD + n0 + l16;
        const float* brow1 = W + (size_t)(k0 + 2 * lhalf + 1) * HD + n0 + l16;

        v2f b0; b0.x = brow0[ 0]; b0.y = brow1[ 0];
        v2f b1; b1.x = brow0[16]; b1.y = brow1[16];
        v2f b2; b2.x = brow0[32]; b2.y = brow1[32];
        v2f b3; b3.x = brow0[48]; b3.y = brow1[48];

        acc0 = __builtin_amdgcn_wmma_f32_16x16x4_f32(false, a, false, b0, (short)0, acc0, false, false);
        acc1 = __builtin_amdgcn_wmma_f32_16x16x4_f32(false, a, false, b1, (short)0, acc1, false, false);
        acc2 = __builtin_amdgcn_wmma_f32_16x16x4_f32(false, a, false, b2, (short)0, acc2, false, false);
        acc3 = __builtin_amdgcn_wmma_f32_16x16x4_f32(false, a, false, b3, (short)0, acc3, false, false);
    }

    // D layout: lane l, reg r -> (M = r + 8*(l>=16), N = l%16)
    #pragma unroll
    for (int r = 0; r < 8; ++r) {
        const size_t row = (size_t)(m0 + r + 8 * lhalf) * HD + n0 + l16;
        feat[row +  0] = acc0[r];
        feat[row + 16] = acc1[r];
        feat[row + 32] = acc2[r];
        feat[row + 48] = acc3[r];
    }
}

// ---------------------------------------------------------------------------
// Kernel 2: el[n,h] = sum_d feat[n,h,d]*attn_l[h,d]; same for er.
// One 256-thread block per node; each wave32 covers exactly one head (D=32).
// ---------------------------------------------------------------------------
__global__ __launch_bounds__(256)
void head_proj(const float* __restrict__ feat,
               const float* __restrict__ attn_l,
               const float* __restrict__ attn_r,
               float* __restrict__ el, float* __restrict__ er) {
    const int n = blockIdx.x;
    const int t = threadIdx.x;               // 0..255 == h*32 + d
    const float f = feat[(size_t)n * HD + t];
    float vl = f * attn_l[t];
    float vr = f * attn_r[t];
    #pragma unroll
    for (int off = 16; off > 0; off >>= 1) {
        vl += __shfl_xor(vl, off, 32);
        vr += __shfl_xor(vr, off, 32);
    }
    if ((t & 31) == 0) {
        el[n * HEADS + (t >> 5)] = vl;
        er[n * HEADS + (t >> 5)] = vr;
    }
}

// ---------------------------------------------------------------------------
// Kernel 3: init segment buffers (mx = -inf, denom = 0).
// ---------------------------------------------------------------------------
__global__ void seg_init(float* __restrict__ mx, float* __restrict__ denom, int n) {
    int i = blockIdx.x * blockDim.x + threadIdx.x;
    if (i < n) { mx[i] = -INFINITY; denom[i] = 0.0f; }
}

__device__ inline void atomicMaxF(float* addr, float val) {
    int cur = __float_as_int(*addr);
    while (__int_as_float(cur) < val) {
        int assumed = cur;
        cur = atomicCAS((int*)addr, assumed, __float_as_int(val));
        if (cur == assumed) break;
    }
}

// ---------------------------------------------------------------------------
// Kernel 4: edge logits e = leaky_relu(el[src]+er[dst]) and segment max.
// ---------------------------------------------------------------------------
__global__ void edge_logits(const float* __restrict__ el,
                            const float* __restrict__ er,
                            const int* __restrict__ src,
                            const int* __restrict__ dst,
                            float* __restrict__ elog,
                            float* __restrict__ mx) {
    int i = blockIdx.x * blockDim.x + threadIdx.x;   // edge*HEADS + h
    if (i >= N_EDGES * HEADS) return;
    const int e = i >> 3;
    const int hh = i & 7;
    const int s = src[e], d = dst[e];
    float x = el[s * HEADS + hh] + er[d * HEADS + hh];
    x = (x > 0.0f) ? x : NEG_SLOPE * x;
    elog[i] = x;
    atomicMaxF(&mx[d * HEADS + hh], x);
}

// ---------------------------------------------------------------------------
// Kernel 5: ee = exp(e - mx[dst]); denom[dst] += ee (in-place over elog).
// ---------------------------------------------------------------------------
__global__ void edge_exp(const int* __restrict__ dst,
                         float* __restrict__ elog,
                         const float* __restrict__ mx,
                         float* __restrict__ denom) {
    int i = blockIdx.x * blockDim.x + threadIdx.x;
    if (i >= N_EDGES * HEADS) return;
    const int e = i >> 3;
    const int hh = i & 7;
    const int d = dst[e];
    float v = __expf(elog[i] - mx[d * HEADS + hh]);
    elog[i] = v;
    atomicAdd(&denom[d * HEADS + hh], v);
}

// ---------------------------------------------------------------------------
// Kernel 6: out[dst] += feat[src] * alpha   (one 256-thread block per edge).
// ---------------------------------------------------------------------------
__global__ __launch_bounds__(256)
void scatter_msg(const float* __restrict__ feat,
                 const int* __restrict__ src,
                 const int* __restrict__ dst,
                 const float* __restrict__ ee,
                 const float* __restrict__ denom,
                 float* __restrict__ out) {
    const int e = blockIdx.x;
    const int t = threadIdx.x;       // h*32 + d
    const int hh = t >> 5;
    const int s = src[e], d = dst[e];
    const float alpha = ee[e * HEADS + hh] / denom[d * HEADS + hh];
    atomicAdd(&out[(size_t)d * HD + t], feat[(size_t)s * HD + t] * alpha);
}

// ---------------------------------------------------------------------------
// Kernel 7: out = elu(out + bias), in place. (Semantic attention stage is an
// exact identity: softmax over a single meta-path == 1.0, so final = elu(rst).)
// ---------------------------------------------------------------------------
__global__ void elu_bias(float* __restrict__ out, const float* __restrict__ bias, int n) {
    int i = blockIdx.x * blockDim.x + threadIdx.x;
    if (i >= n) return;
    float x = out[i] + bias[i & (HD - 1)];
    out[i] = (x > 0.0f) ? x : (__expf(x) - 1.0f);
}

// ---------------------------------------------------------------------------
extern "C" void kernel_launch(void* const* d_in, const int* in_sizes, int n_in,
                              void* d_out, int out_size, void* d_ws, size_t ws_size,
                              hipStream_t stream) {
    const float* h      = (const float*)d_in[0];
    const float* W_fc   = (const float*)d_in[1];
    const float* attn_l = (const float*)d_in[2];
    const float* attn_r = (const float*)d_in[3];
    const float* bias   = (const float*)d_in[4];
    // d_in[5..7] = sem_W1/sem_b1/sem_W2: mathematically identity on output, skipped.
    const int* src = (const int*)d_in[8];
    const int* dst = (const int*)d_in[9];
    float* out = (float*)d_out;

    // Workspace carve-up (floats)
    float* ws    = (float*)d_ws;
    float* feat  = ws;                                    // N*HD      = 25.6M
    float* el    = feat  + (size_t)N_NODES * HD;          // N*H
    float* er    = el    + (size_t)N_NODES * HEADS;       // N*H
    float* mx    = er    + (size_t)N_NODES * HEADS;       // N*H
    float* denom = mx    + (size_t)N_NODES * HEADS;       // N*H
    float* elog  = denom + (size_t)N_NODES * HEADS;       // E*H (reused as ee)

    // 1. GEMM (WMMA f32)
    dim3 ggrid(N_NODES / 16, HD / 64);
    gemm_feat_wmma<<<ggrid, 32, 0, stream>>>(h, W_fc, feat);

    // 2. per-head projections
    head_proj<<<N_NODES, 256, 0, stream>>>(feat, attn_l, attn_r, el, er);

    // 3. init segment buffers + zero the accumulation output
    {
        int n = N_NODES * HEADS;
        seg_init<<<(n + 255) / 256, 256, 0, stream>>>(mx, denom, n);
        hipMemsetAsync(d_out, 0, (size_t)out_size * sizeof(float), stream);
    }

    // 4. edge logits + segment max
    {
        int n = N_EDGES * HEADS;
        edge_logits<<<(n + 255) / 256, 256, 0, stream>>>(el, er, src, dst, elog, mx);
        // 5. exp + segment sum
        edge_exp<<<(n + 255) / 256, 256, 0, stream>>>(dst, elog, mx, denom);
    }

    // 6. weighted message scatter
    scatter_msg<<<N_EDGES, 256, 0, stream>>>(feat, src, dst, elog, denom, out);

    // 7. bias + ELU in place (semantic stage == identity)
    {
        int n = N_NODES * HD;
        elu_bias<<<(n + 255) / 256, 256, 0, stream>>>(out, bias, n);
    }
}